// HoG_4947802325733
// MI455X (gfx1250) — compile-verified
//
#include <hip/hip_runtime.h>

typedef __attribute__((ext_vector_type(2))) float v2f;
typedef __attribute__((ext_vector_type(8))) float v8f;

#define HOG_B   8
#define HOG_C   3
#define HOG_H   1024
#define HOG_W   1024
#define NBINS   9
#define CELL    8
#define TILE_W  128                 // columns per block -> 16 cells
#define HCELLS  (HOG_H / CELL)      // 128
#define WCELLS  (HOG_W / CELL)      // 128
#define LDS_PITCH 17                // 16 padded bins + 1 to kill bank conflicts

__global__ __launch_bounds__(TILE_W)
void hog_cells_kernel(const float* __restrict__ img, float* __restrict__ out)
{
    // per-column vote table: [128 columns][16 bins (9 real + 7 zero)] padded
    __shared__ float colvote[TILE_W * LDS_PITCH];

    const int x  = threadIdx.x;          // column within tile, 0..127
    const int tx = blockIdx.x;           // tile along W, 0..7
    const int cy = blockIdx.y;           // cell row, 0..127
    const int b  = blockIdx.z;           // batch
    const int xg = tx * TILE_W + x;      // global column
    const int y0 = cy * CELL;

    // zero the vote table; bins 9..15 stay zero -> clean WMMA B operand
#pragma unroll
    for (int k = 0; k < LDS_PITCH; ++k)
        colvote[x * LDS_PITCH + k] = 0.0f;
    __syncthreads();

    const float* base = img + (size_t)b * HOG_C * HOG_H * HOG_W;

    // reflect-pad(1) + central difference => gradient is exactly 0 on borders
    const bool xedge = (xg == 0) | (xg == HOG_W - 1);

    // Rolling 2-deep window per channel: prev[c][r&1] holds img[c][y-1][xg]
    // at iteration r (val(y+1) of row r is val(y-1) of row r+2).
    float prev[HOG_C][2];
#pragma unroll
    for (int c = 0; c < HOG_C; ++c) {
        const float* pc = base + (size_t)c * HOG_H * HOG_W + xg;
        const int ym = (y0 > 0) ? (y0 - 1) : 0;   // clamped; dead value at y==0
        prev[c][0] = pc[(size_t)ym * HOG_W];
        prev[c][1] = pc[(size_t)y0 * HOG_W];
    }

    float* myrow = &colvote[x * LDS_PITCH];

#pragma unroll
    for (int r = 0; r < CELL; ++r) {
        const int y = y0 + r;
        const bool yedge = (y == 0) | (y == HOG_H - 1);
        const int yp = (y < HOG_H - 1) ? (y + 1) : (HOG_H - 1); // clamped load

        float bestM = -1.0f, bgx = 0.0f, bgy = 0.0f;
#pragma unroll
        for (int c = 0; c < HOG_C; ++c) {
            const float* pr   = base + ((size_t)c * HOG_H + y) * HOG_W + xg;
            const float below = base[((size_t)c * HOG_H + yp) * HOG_W + xg];
            const float gx = xedge ? 0.0f : (pr[1] - pr[-1]);
            const float gy = yedge ? 0.0f : (below - prev[c][r & 1]);
            prev[c][r & 1] = below;              // becomes val(y-1) at r+2
            const float m2 = fmaf(gx, gx, fmaf(gy, gy, 1e-12f));
            if (m2 > bestM) { bestM = m2; bgx = gx; bgy = gy; } // first-max wins
        }
        const float mag = sqrtf(bestM);
        float ang = atan2f(bgy, bgx);                    // (-pi, pi]
        if (ang < 0.0f) ang += 3.14159265358979323846f;  // mod pi -> [0, pi]

        const float bb  = ang * (NBINS / 3.14159265358979323846f) - 0.5f;
        const float fb0 = floorf(bb);
        const float w1  = bb - fb0;
        const float w0  = 1.0f - w1;
        int i0 = (int)fb0;                     // -1 .. 8
        if (i0 < 0) i0 = NBINS - 1;            // mod 9
        int i1 = i0 + 1;
        if (i1 == NBINS) i1 = 0;

        // Two native LDS float atomics replace a 9-wide select scatter
        // (ds_add_f32, non-returning; DS pipe co-executes with VALU).
        atomicAdd(&myrow[i0], mag * w0);
        atomicAdd(&myrow[i1], mag * w1);
    }

    __syncthreads();

    // Wave 0: hist(16x9) = S(16x128) x colvote(128x16) via 32 chained
    // V_WMMA_F32_16X16X4_F32. Each K-slice of 4 columns lies in one cell,
    // so A is (M == t>>1) generated in registers.
    if (threadIdx.x < 32) {
        const int lane = threadIdx.x;
        const int n    = lane & 15;        // N (bins) for B/D, M (cells) for A
        const int hi   = lane >> 4;        // lane half
        const int kofs = hi * 2;           // K offset inside a 4-wide slice

        v8f d = {0.f, 0.f, 0.f, 0.f, 0.f, 0.f, 0.f, 0.f};
#pragma unroll
        for (int t = 0; t < 32; ++t) {
            const float sel = (n == (t >> 1)) ? 1.0f : 0.0f;  // A[m][k] of slice t
            v2f a; a.x = sel; a.y = sel;
            v2f bm;
            bm.x = colvote[(4 * t + kofs + 0) * LDS_PITCH + n];
            bm.y = colvote[(4 * t + kofs + 1) * LDS_PITCH + n];
            // D = A(16x4) * B(4x16) + D
            d = __builtin_amdgcn_wmma_f32_16x16x4_f32(false, a, false, bm,
                                                      (short)0, d, false, false);
        }

        // D layout: VGPR r, lane -> cell m = r + 8*hi, bin n = lane&15
        if (n < NBINS) {
            const int cellBase = tx * 16 + hi * 8;
            float* o = out + (((size_t)b * HCELLS + cy) * WCELLS + cellBase) * NBINS + n;
#pragma unroll
            for (int r = 0; r < 8; ++r)
                o[(size_t)r * NBINS] = d[r];
        }
    }
}

extern "C" void kernel_launch(void* const* d_in, const int* in_sizes, int n_in,
                              void* d_out, int out_size, void* d_ws, size_t ws_size,
                              hipStream_t stream)
{
    (void)in_sizes; (void)n_in; (void)d_ws; (void)ws_size; (void)out_size;
    const float* img = (const float*)d_in[0];
    float* out = (float*)d_out;

    dim3 grid(HOG_W / TILE_W, HCELLS, HOG_B);   // (8, 128, 8)
    hog_cells_kernel<<<grid, TILE_W, 0, stream>>>(img, out);
}